// ArcDecoder_43662637531154
// MI455X (gfx1250) — compile-verified
//
#include <hip/hip_runtime.h>

typedef float v2f __attribute__((ext_vector_type(2)));
typedef float v8f __attribute__((ext_vector_type(8)));

#define HDIM 64
#define WPAD 68            // padded LDS row stride (floats) -> conflict-free
#define EPS_LN 1e-5f

// ---------------------------------------------------------------------------
// Post-GEMM epilogue on WMMA C/D-layout accumulators:
//   acc[nt][v] holds element (row M = v + 8*hi, col n = nt*16 + l16).
//   bias + relu, then per-row LayerNorm via width-16 shuffle reduction.
// ---------------------------------------------------------------------------
__device__ __forceinline__ void bias_relu_ln(v8f acc[4],
                                             const float* __restrict__ bias,
                                             const float* __restrict__ ng,
                                             const float* __restrict__ nb,
                                             int l16) {
    float bsc[4], gsc[4], nbs[4];
#pragma unroll
    for (int nt = 0; nt < 4; ++nt) {
        int n = nt * 16 + l16;
        bsc[nt] = bias[n]; gsc[nt] = ng[n]; nbs[nt] = nb[n];
    }
#pragma unroll
    for (int nt = 0; nt < 4; ++nt)
#pragma unroll
        for (int v = 0; v < 8; ++v) {
            float x = acc[nt][v] + bsc[nt];
            acc[nt][v] = x > 0.0f ? x : 0.0f;
        }
#pragma unroll
    for (int v = 0; v < 8; ++v) {              // one row per (v, lane-half)
        float s = 0.0f, q = 0.0f;
#pragma unroll
        for (int nt = 0; nt < 4; ++nt) { float x = acc[nt][v]; s += x; q += x * x; }
#pragma unroll
        for (int m = 1; m < 16; m <<= 1) {     // reduce within each 16-lane half
            s += __shfl_xor(s, m, 16);
            q += __shfl_xor(q, m, 16);
        }
        float mu  = s * (1.0f / HDIM);
        float var = q * (1.0f / HDIM) - mu * mu;
        float rs  = rsqrtf(var + EPS_LN);
#pragma unroll
        for (int nt = 0; nt < 4; ++nt)
            acc[nt][v] = (acc[nt][v] - mu) * rs * gsc[nt] + nbs[nt];
    }
}

// ---------------------------------------------------------------------------
// Node stage: per node i (0..N):
//   zz  = LN(i==0 ? root_w+root_b : z[i-1])
//   X1  = LN(relu(zz @ W1^T + b1));  U1 = X1 @ Bw     (bilinear folded in)
//   X2  = LN(relu(zz @ W2^T + b2))
// One wave = 16 rows. 3 GEMMs via V_WMMA_F32_16X16X4_F32.
// ---------------------------------------------------------------------------
__global__ __launch_bounds__(128)
void node_transform_kernel(const float* __restrict__ z,
                           const float* __restrict__ root_w,
                           const float* __restrict__ root_b,
                           const float* __restrict__ w1, const float* __restrict__ b1,
                           const float* __restrict__ w2, const float* __restrict__ b2,
                           const float* __restrict__ bw,
                           const float* __restrict__ ng, const float* __restrict__ nb,
                           float* __restrict__ U1, float* __restrict__ X2,
                           int nodes) {
    __shared__ float lds_w[2][HDIM * WPAD];        // slot0: W1 then Bw; slot1: W2
    __shared__ float lds_tile[4][16 * WPAD];       // per-wave 16x64 staging tile

    const int t    = threadIdx.x;
    const int wave = t >> 5;
    const int lane = t & 31;
    const int hi   = lane >> 4;                    // lane half (K/M split)
    const int l16  = lane & 15;

    // stage W1 / W2 into LDS (padded stride)
#pragma unroll 4
    for (int i = 0; i < 32; ++i) {
        int idx = t + i * 128;
        int r = idx >> 6, c = idx & 63;
        lds_w[0][r * WPAD + c] = w1[idx];
        lds_w[1][r * WPAD + c] = w2[idx];
    }

    const int rowBase = blockIdx.x * 64 + wave * 16;
    float* tile = lds_tile[wave];

    // ---- load 16 input rows, LayerNorm in registers (2 lanes per row) ----
    {
        int r     = lane >> 1;
        int halfc = (lane & 1) * 32;
        int node  = rowBase + r;
        int cn    = node < nodes ? node : nodes - 1;
        float x[32];
        if (cn == 0) {
#pragma unroll
            for (int c = 0; c < 32; ++c) x[c] = root_w[halfc + c] + root_b[halfc + c];
        } else {
            const float* src = z + (size_t)(cn - 1) * HDIM + halfc;
#pragma unroll
            for (int c = 0; c < 32; c += 4) {
                float4 v = *(const float4*)(src + c);
                x[c] = v.x; x[c + 1] = v.y; x[c + 2] = v.z; x[c + 3] = v.w;
            }
        }
        float s = 0.0f, q = 0.0f;
#pragma unroll
        for (int c = 0; c < 32; ++c) { s += x[c]; q += x[c] * x[c]; }
        s += __shfl_xor(s, 1, 32);
        q += __shfl_xor(q, 1, 32);
        float mu  = s * (1.0f / HDIM);
        float var = q * (1.0f / HDIM) - mu * mu;
        float rs  = rsqrtf(var + EPS_LN);
        float* dst = tile + r * WPAD + halfc;
#pragma unroll
        for (int c = 0; c < 32; ++c)
            dst[c] = (x[c] - mu) * rs * ng[halfc + c] + nb[halfc + c];
    }
    __syncthreads();

    // ---- GEMM1 & GEMM2 (shared A-fragments): zz @ W1^T, zz @ W2^T ----
    v8f acc1[4], acc2[4];
    const float* arow = tile + l16 * WPAD;
#pragma unroll
    for (int nt = 0; nt < 4; ++nt) {
        v8f c1 = {0, 0, 0, 0, 0, 0, 0, 0};
        v8f c2 = {0, 0, 0, 0, 0, 0, 0, 0};
        const int n = nt * 16 + l16;
#pragma unroll
        for (int k = 0; k < 16; ++k) {
            const int kc = 4 * k + 2 * hi;                              // K column pair
            v2f a  = *(const v2f*)(arow + kc);                          // A[M=l16][kc,kc+1]
            v2f bA = *(const v2f*)(&lds_w[0][n * WPAD + kc]);           // W1[n][kc,kc+1]
            v2f bB = *(const v2f*)(&lds_w[1][n * WPAD + kc]);           // W2[n][kc,kc+1]
            c1 = __builtin_amdgcn_wmma_f32_16x16x4_f32(false, a, false, bA,
                                                       (short)0, c1, false, false);
            c2 = __builtin_amdgcn_wmma_f32_16x16x4_f32(false, a, false, bB,
                                                       (short)0, c2, false, false);
        }
        acc1[nt] = c1;
        acc2[nt] = c2;
    }

    // ---- epilogues: bias + relu + LayerNorm on both paths ----
    bias_relu_ln(acc1, b1, ng, nb, l16);
    bias_relu_ln(acc2, b2, ng, nb, l16);

    // X2 path: store normalized rows straight to global (C/D layout scatter,
    // 16 lanes -> 64 contiguous bytes per store group)
#pragma unroll
    for (int nt = 0; nt < 4; ++nt)
#pragma unroll
        for (int v = 0; v < 8; ++v) {
            int node = rowBase + v + 8 * hi;
            if (node < nodes)
                X2[(size_t)node * HDIM + nt * 16 + l16] = acc2[nt][v];
        }

    // X1 path: transpose via LDS (C/D layout -> row major) for next A-frags
#pragma unroll
    for (int nt = 0; nt < 4; ++nt)
#pragma unroll
        for (int v = 0; v < 8; ++v)
            tile[(v + 8 * hi) * WPAD + nt * 16 + l16] = acc1[nt][v];
    __syncthreads();

    // stage bilinear matrix into slot0 (W1 no longer needed)
#pragma unroll 4
    for (int i = 0; i < 32; ++i) {
        int idx = t + i * 128;
        int r = idx >> 6, c = idx & 63;
        lds_w[0][r * WPAD + c] = bw[idx];
    }
    __syncthreads();

    // ---- GEMM3: U1 = X1 @ Bw   (B[k][n] = bw[k][n], row-major in LDS) ----
#pragma unroll
    for (int nt = 0; nt < 4; ++nt) {
        v8f c = {0, 0, 0, 0, 0, 0, 0, 0};
        const int n = nt * 16 + l16;
#pragma unroll
        for (int k = 0; k < 16; ++k) {
            const int kc = 4 * k + 2 * hi;
            v2f a = *(const v2f*)(arow + kc);
            v2f b;
            b.x = lds_w[0][kc * WPAD + n];
            b.y = lds_w[0][(kc + 1) * WPAD + n];
            c = __builtin_amdgcn_wmma_f32_16x16x4_f32(false, a, false, b,
                                                      (short)0, c, false, false);
        }
#pragma unroll
        for (int v = 0; v < 8; ++v) {
            int node = rowBase + v + 8 * hi;
            if (node < nodes)
                U1[(size_t)node * HDIM + n] = c[v];
        }
    }
}

// ---------------------------------------------------------------------------
// Edge stage: score[e] = dot(U1[arc0], X2[arc1]) + bil_b.
// 8 lanes per edge -> each 256B row read as coalesced float4s (L2 resident),
// width-8 shuffle reduction.
// ---------------------------------------------------------------------------
__global__ __launch_bounds__(256)
void edge_score_kernel(const int* __restrict__ arcs,
                       const float* __restrict__ U1,
                       const float* __restrict__ X2,
                       const float* __restrict__ bilb,
                       float* __restrict__ out, int E) {
    int t   = blockIdx.x * 256 + threadIdx.x;
    int e   = t >> 3;
    int sub = t & 7;
    if (e >= E) return;
    int a = arcs[2 * (size_t)e + 0];
    int b = arcs[2 * (size_t)e + 1];
    const float4* pa = (const float4*)(U1 + (size_t)a * HDIM) + sub * 2;
    const float4* pb = (const float4*)(X2 + (size_t)b * HDIM) + sub * 2;
    float4 a0 = pa[0], a1 = pa[1];
    float4 b0 = pb[0], b1 = pb[1];
    float s = a0.x * b0.x + a0.y * b0.y + a0.z * b0.z + a0.w * b0.w
            + a1.x * b1.x + a1.y * b1.y + a1.z * b1.z + a1.w * b1.w;
    s += __shfl_xor(s, 1, 8);
    s += __shfl_xor(s, 2, 8);
    s += __shfl_xor(s, 4, 8);
    if (sub == 0) out[e] = s + bilb[0];
}

extern "C" void kernel_launch(void* const* d_in, const int* in_sizes, int n_in,
                              void* d_out, int out_size, void* d_ws, size_t ws_size,
                              hipStream_t stream) {
    const float* z      = (const float*)d_in[0];
    const int*   arcs   = (const int*)d_in[1];
    const float* root_w = (const float*)d_in[2];
    const float* root_b = (const float*)d_in[3];
    const float* w1     = (const float*)d_in[4];
    const float* b1     = (const float*)d_in[5];
    const float* w2     = (const float*)d_in[6];
    const float* b2     = (const float*)d_in[7];
    const float* bw     = (const float*)d_in[8];
    const float* bilb   = (const float*)d_in[9];
    const float* ng     = (const float*)d_in[10];
    const float* nb     = (const float*)d_in[11];

    const int N     = in_sizes[0] / HDIM;
    const int E     = in_sizes[1] / 2;
    const int nodes = N + 1;

    float* U1 = (float*)d_ws;
    float* X2 = U1 + (size_t)nodes * HDIM;

    int nblocks = (nodes + 63) / 64;
    node_transform_kernel<<<nblocks, 128, 0, stream>>>(
        z, root_w, root_b, w1, b1, w2, b2, bw, ng, nb, U1, X2, nodes);

    int eblocks = (E * 8 + 255) / 256;
    edge_score_kernel<<<eblocks, 256, 0, stream>>>(
        arcs, U1, X2, bilb, (float*)d_out, E);
}